// PatchNet_8761733283898
// MI455X (gfx1250) — compile-verified
//
#include <hip/hip_runtime.h>
#include <hip/hip_bf16.h>
#include <math.h>

// ---- problem constants (from reference) ----
#define CC 384
#define DD 768
#define TT 32
#define NN 196
#define BB 16
#define KK 16
#define NS 500
#define NC (NN * CC)   // 75264 flattened frame width

typedef __attribute__((ext_vector_type(2))) float v2f;
typedef __attribute__((ext_vector_type(8))) float v8f;

__device__ __forceinline__ float gelu_exact(float v) {
  // matches jax.nn.gelu(approximate=False)
  return 0.5f * v * (1.0f + erff(v * 0.7071067811865475f));
}

// ---------------------------------------------------------------------------
// Kernel 1: fused frame pooling (mean+max over N) + LayerNorm over D=768.
// One block per (b,t) row, 384 threads (one per channel), block reduction
// for mu/var. Streams x exactly once (154 MB) -> h0 (512x768).
// ---------------------------------------------------------------------------
__global__ void pool_ln_kernel(const float* __restrict__ x,
                               const float* __restrict__ ln_g,
                               const float* __restrict__ ln_b,
                               float* __restrict__ h0) {
  int bt = blockIdx.x;            // b*T + t
  int c  = threadIdx.x;           // 0..383
  const float* base = x + (size_t)bt * NC + c;
  float s = 0.f, mx = -3.402823466e38f;
  for (int n = 0; n < NN; ++n) {
    float v = base[(size_t)n * CC];   // coalesced across c
    s += v;
    mx = fmaxf(mx, v);
  }
  float mean = s * (1.0f / (float)NN);

  __shared__ float red1[CC];
  __shared__ float red2[CC];
  red1[c] = mean + mx;
  red2[c] = mean * mean + mx * mx;
  __syncthreads();
  if (c < 128) { red1[c] += red1[c + 256]; red2[c] += red2[c + 256]; }
  __syncthreads();
  for (int off = 128; off > 0; off >>= 1) {
    if (c < off) { red1[c] += red1[c + off]; red2[c] += red2[c + off]; }
    __syncthreads();
  }
  float mu   = red1[0] * (1.0f / (float)DD);
  float var  = red2[0] * (1.0f / (float)DD) - mu * mu;   // biased, like nn.LayerNorm
  float rstd = rsqrtf(var + 1e-5f);

  size_t o = (size_t)bt * DD;
  h0[o + c]      = (mean - mu) * rstd * ln_g[c]      + ln_b[c];
  h0[o + CC + c] = (mx   - mu) * rstd * ln_g[CC + c] + ln_b[CC + c];
}

// ---------------------------------------------------------------------------
// Generic fp32 WMMA GEMM: C = [GELU](A[M,K] @ B[K,N] + bias[N])
// One wave per 16x16 output tile, K-loop of V_WMMA_F32_16X16X4_F32.
// Grids are exact (tiles % waves_per_block == 0) so EXEC is all-ones.
// A frag (16x4): M in lanes 0-15, K-pair selected by lane-half.
// B frag (4x16): N in lanes 0-15, same K-pair split. C/D per ISA 7.12.2.
// ---------------------------------------------------------------------------
template <int APPLY_GELU>
__global__ void wmma_gemm_kernel(const float* __restrict__ A,
                                 const float* __restrict__ B,
                                 const float* __restrict__ bias,
                                 float* __restrict__ Cout,
                                 int M, int Nn, int Kd) {
  int wave   = blockIdx.x * (blockDim.x >> 5) + (threadIdx.x >> 5);
  int tilesN = Nn >> 4;
  int tm = (wave / tilesN) << 4;
  int tn = (wave % tilesN) << 4;
  int lane = threadIdx.x & 31;
  int l15  = lane & 15;
  int koff = (lane >> 4) << 1;          // 0 or 2

  const float* Arow = A + (size_t)(tm + l15) * Kd;
  const float* Bcol = B + tn + l15;

  v8f acc = {};
  for (int k = 0; k < Kd; k += 4) {
    v2f a, b;
    a.x = Arow[k + koff];
    a.y = Arow[k + koff + 1];
    b.x = Bcol[(size_t)(k + koff) * Nn];
    b.y = Bcol[(size_t)(k + koff + 1) * Nn];
    acc = __builtin_amdgcn_wmma_f32_16x16x4_f32(false, a, false, b,
                                                (short)0, acc, false, false);
  }

  int col   = tn + l15;
  int mBase = tm + ((lane >> 4) << 3);
  float bv  = bias ? bias[col] : 0.f;
#pragma unroll
  for (int i = 0; i < 8; ++i) {
    float v = acc[i] + bv;
    if (APPLY_GELU) v = gelu_exact(v);
    Cout[(size_t)(mBase + i) * Nn + col] = v;
  }
}

// ---------------------------------------------------------------------------
// Kernel 3: global pool over T of the second half of h1 -> g (B x 384)
// ---------------------------------------------------------------------------
__global__ void gpool_kernel(const float* __restrict__ h1, float* __restrict__ g) {
  int b = blockIdx.x, c = threadIdx.x;
  float s = 0.f;
  for (int t = 0; t < TT; ++t)
    s += h1[(size_t)(b * TT + t) * DD + CC + c];
  g[b * CC + c] = s * (1.0f / (float)TT);
}

// Kernel 4: h2 = concat(h1[:, :384], broadcast g)
__global__ void concat_kernel(const float* __restrict__ h1,
                              const float* __restrict__ g,
                              float* __restrict__ h2) {
  int bt = blockIdx.x, c = threadIdx.x;
  int b = bt >> 5;
  size_t o = (size_t)bt * DD;
  h2[o + c]      = h1[o + c];
  h2[o + CC + c] = g[b * CC + c];
}

// ---------------------------------------------------------------------------
// Kernel 7: final 192-dim dot product per (b,t) row; one wave per row.
// ---------------------------------------------------------------------------
__global__ void score_kernel(const float* __restrict__ h4,
                             const float* __restrict__ w3,
                             const float* __restrict__ b3,
                             float* __restrict__ scores) {
  int bt = blockIdx.x;
  int lane = threadIdx.x;
  const float* row = h4 + (size_t)bt * 192;
  float p = 0.f;
#pragma unroll
  for (int j = 0; j < 6; ++j)
    p += row[lane + 32 * j] * w3[lane + 32 * j];
#pragma unroll
  for (int off = 16; off > 0; off >>= 1)
    p += __shfl_down(p, off, 32);
  if (lane == 0) scores[bt] = p + b3[0];
}

// ---------------------------------------------------------------------------
// Kernel 8: min-max norm + perturbed top-K indicators.
// One block per batch b (8 waves); each wave owns samples s = wid + 8*j.
// T=32 fits one value per lane of a wave32. Rank via 32 shuffles with
// jax-top_k stable tie-break (lower index wins); selected if rank < 16.
// k-position = # selected lanes with smaller t (shuffle prefix count).
// Per-wave private LDS histograms -> deterministic reduction, no atomics.
// ---------------------------------------------------------------------------
__global__ void topk_indicator_kernel(const float* __restrict__ scores,
                                      const float* __restrict__ noise,
                                      float* __restrict__ ind) {
  __shared__ float sn[TT];
  __shared__ float hist[8][KK * TT];
  __shared__ float slo, shi;

  int b = blockIdx.x;
  int tid = threadIdx.x;
  int lane = tid & 31;
  int wid  = tid >> 5;   // 0..7

  for (int i = tid; i < 8 * KK * TT; i += blockDim.x)
    (&hist[0][0])[i] = 0.f;

  if (tid == 0) {
    float lo = scores[b * TT], hi = lo;
    for (int t = 1; t < TT; ++t) {
      float v = scores[b * TT + t];
      lo = fminf(lo, v);
      hi = fmaxf(hi, v);
    }
    slo = lo; shi = hi;
  }
  __syncthreads();
  if (tid < TT)
    sn[tid] = (scores[b * TT + tid] - slo) / (shi - slo + 1e-5f);
  __syncthreads();

  float base = sn[lane];
  for (int s = wid; s < NS; s += 8) {           // wave-uniform bounds
    float v = base + noise[((size_t)b * NS + s) * TT + lane] * 0.05f;
    int rank = 0;
    for (int j = 0; j < 32; ++j) {
      float o = __shfl(v, j, 32);
      rank += (o > v) || (o == v && j < lane);
    }
    int sel = (rank < KK) ? 1 : 0;
    int kpos = 0;
    for (int j = 0; j < 32; ++j) {
      int sj = __shfl(sel, j, 32);
      kpos += (j < lane) ? sj : 0;
    }
    if (sel) hist[wid][kpos * TT + lane] += 1.f;  // lane-unique address
  }
  __syncthreads();

  for (int i = tid; i < KK * TT; i += blockDim.x) {
    float s = 0.f;
#pragma unroll
    for (int w = 0; w < 8; ++w) s += hist[w][i];
    ind[(size_t)b * KK * TT + i] = s * (1.0f / (float)NS);
  }
}

// ---------------------------------------------------------------------------
// Kernel 9: gather einsum as per-batch WMMA GEMM:
//   out[b] (16 x 75264) = ind[b] (16 x 32) @ xflat[b] (32 x 75264)
// B-fragment loads are unit-stride in the flattened (n*C + c) axis -> fully
// coalesced. 8 f32 WMMAs per tile, 4704 tiles per batch.
// ---------------------------------------------------------------------------
__global__ void gather_wmma_kernel(const float* __restrict__ ind,
                                   const float* __restrict__ x,
                                   float* __restrict__ out) {
  int b    = blockIdx.y;
  int wave = blockIdx.x * 4 + (threadIdx.x >> 5);   // 0..4703
  int tn   = wave << 4;
  int lane = threadIdx.x & 31;
  int l15  = lane & 15;
  int koff = (lane >> 4) << 1;

  const float* A = ind + (size_t)b * KK * TT;       // 16x32 row-major
  const float* B = x + (size_t)b * TT * NC;         // 32x75264 row-major

  v8f acc = {};
#pragma unroll
  for (int k = 0; k < TT; k += 4) {
    v2f a, bb;
    a.x  = A[l15 * TT + k + koff];
    a.y  = A[l15 * TT + k + koff + 1];
    bb.x = B[(size_t)(k + koff) * NC + tn + l15];
    bb.y = B[(size_t)(k + koff + 1) * NC + tn + l15];
    acc = __builtin_amdgcn_wmma_f32_16x16x4_f32(false, a, false, bb,
                                                (short)0, acc, false, false);
  }

  int col   = tn + l15;
  int mBase = (lane >> 4) << 3;
  float* O  = out + (size_t)b * KK * NC;
#pragma unroll
  for (int i = 0; i < 8; ++i)
    O[(size_t)(mBase + i) * NC + col] = acc[i];
}

// ---------------------------------------------------------------------------
extern "C" void kernel_launch(void* const* d_in, const int* in_sizes, int n_in,
                              void* d_out, int out_size, void* d_ws, size_t ws_size,
                              hipStream_t stream) {
  const float* x     = (const float*)d_in[0];
  const float* noise = (const float*)d_in[1];
  const float* ln_g  = (const float*)d_in[2];
  const float* ln_b  = (const float*)d_in[3];
  const float* w_in  = (const float*)d_in[4];
  const float* b_in  = (const float*)d_in[5];
  const float* w1    = (const float*)d_in[6];
  const float* b1    = (const float*)d_in[7];
  const float* w2    = (const float*)d_in[8];
  const float* b2    = (const float*)d_in[9];
  const float* w3    = (const float*)d_in[10];
  const float* b3    = (const float*)d_in[11];
  float* out = (float*)d_out;

  // workspace layout (floats), ~6 MB total
  float* ws = (float*)d_ws;
  float* h0 = ws;                       // 512*768  LN output
  float* h1 = h0 + 512 * DD;            // 512*768  GELU(h0 @ w_in)
  float* h2 = h1 + 512 * DD;            // 512*768  concat(local, pooled)
  float* gp = h2 + 512 * DD;            // 16*384   T-pooled global half
  float* h3 = gp + BB * CC;             // 512*384
  float* h4 = h3 + 512 * CC;            // 512*192
  float* sc = h4 + 512 * 192;           // 512      raw scores
  float* id = sc + 512;                 // 16*16*32 indicators

  pool_ln_kernel<<<BB * TT, CC, 0, stream>>>(x, ln_g, ln_b, h0);

  // GEMM1: (512x768)@(768x768)+GELU -> 1536 tiles, 4 waves/block
  wmma_gemm_kernel<1><<<(32 * 48) / 4, 128, 0, stream>>>(h0, w_in, b_in, h1, 512, DD, DD);

  gpool_kernel<<<BB, CC, 0, stream>>>(h1, gp);
  concat_kernel<<<BB * TT, CC, 0, stream>>>(h1, gp, h2);

  // GEMM2: (512x768)@(768x384)+GELU -> 768 tiles
  wmma_gemm_kernel<1><<<(32 * 24) / 4, 128, 0, stream>>>(h2, w1, b1, h3, 512, CC, DD);
  // GEMM3: (512x384)@(384x192)+GELU -> 384 tiles
  wmma_gemm_kernel<1><<<(32 * 12) / 4, 128, 0, stream>>>(h3, w2, b2, h4, 512, 192, CC);

  score_kernel<<<BB * TT, 32, 0, stream>>>(h4, w3, b3, sc);
  topk_indicator_kernel<<<BB, 256, 0, stream>>>(sc, noise, id);

  // gather: per-batch (16x32)@(32x75264); 4704 tiles/batch, 4 waves/block
  gather_wmma_kernel<<<dim3(NC / 16 / 4, BB), 128, 0, stream>>>(id, x, out);
}